// EdgeConv_27547920237121
// MI455X (gfx1250) — compile-verified
//
#include <hip/hip_runtime.h>

// EdgeConv for MI455X (gfx1250, wave32, WMMA).
// Pipeline:
//   K1: squared norms + zero BN accumulators
//   K2: KNN via V_WMMA_F32_16X16X4_F32 distance tiles, whole batch staged in 128KB LDS
//   K3: edge-feature x W (6->64) via WMMA, accumulate per-channel sum/sumsq
//   K4: finalize BN scale/shift
//   K5: recompute h via WMMA, apply BN+ReLU, coalesced stores via LDS transpose

typedef float v2f __attribute__((ext_vector_type(2)));
typedef float v8f __attribute__((ext_vector_type(8)));

#define NB   4
#define NPTS 8192
#define NOUT 64
#define KNN  20
#define EPB  (NPTS * KNN)        // edges per batch = 163840
#define TOTE (NB * EPB)          // total edges     = 655360

__device__ __forceinline__ v8f wmma4(v2f a, v2f b, v8f c) {
  // D = A(16x4) * B(4x16) + C(16x16), f32
  return __builtin_amdgcn_wmma_f32_16x16x4_f32(false, a, false, b, (short)0, c,
                                               false, false);
}

// ---------------------------------------------------------------- kernel 1
__global__ __launch_bounds__(256) void sq_kernel(const float* __restrict__ x,
                                                 float* __restrict__ sq,
                                                 float* __restrict__ stats) {
  int i = blockIdx.x * 256 + threadIdx.x;
  if (i < NB * NPTS) {
    float a = x[3 * i], b = x[3 * i + 1], c = x[3 * i + 2];
    sq[i] = fmaf(a, a, fmaf(b, b, c * c));
  }
  if (blockIdx.x == 0 && threadIdx.x < 128) stats[threadIdx.x] = 0.f;  // sum/sumsq
}

// ---------------------------------------------------------------- kernel 2
// One block = 256 queries of one batch (8 waves x 32 queries).
// Whole batch (x,y,z,sq) staged as float4 in LDS (128KB of the 320KB WGP LDS).
__global__ __launch_bounds__(256) void knn_kernel(const float* __restrict__ x,
                                                  const float* __restrict__ sq,
                                                  int* __restrict__ knn_idx) {
  __shared__ float4 xs[NPTS];           // 128 KB
  __shared__ float dtile[8][32][16];    // 16 KB per-wave scratch

  const int b  = blockIdx.x >> 5;           // 32 blocks per batch
  const int qb = (blockIdx.x & 31) * 256;   // query base within batch
  const int w  = threadIdx.x >> 5;
  const int l  = threadIdx.x & 31;
  const int hf = l >> 4;
  const int ln = l & 15;

  const float* xb  = x  + (size_t)b * NPTS * 3;
  const float* sqb = sq + (size_t)b * NPTS;
  for (int j = threadIdx.x; j < NPTS; j += 256) {
    float4 p;
    p.x = xb[3 * j]; p.y = xb[3 * j + 1]; p.z = xb[3 * j + 2]; p.w = sqb[j];
    xs[j] = p;
  }
  __syncthreads();

  // A operands for this wave's 32 query rows (two 16-row tiles), K padded 3->4.
  const int qw = qb + w * 32;
  float4 p0 = xs[qw + ln];
  float4 p1 = xs[qw + 16 + ln];
  v2f a0; a0.x = hf ? p0.z : p0.x; a0.y = hf ? 0.f : p0.y;
  v2f a1; a1.x = hf ? p1.z : p1.x; a1.y = hf ? 0.f : p1.y;
  const float sqI = hf ? p1.w : p0.w;   // owner lane l <-> query qw + l

  float kd[KNN]; int ki[KNN];
#pragma unroll
  for (int t = 0; t < KNN; ++t) { kd[t] = 3.0e38f; ki[t] = 0; }
  float curMax = 3.0e38f; int maxSlot = 0;

  const float* rowp = &dtile[w][l][0];  // this lane's 64B-aligned distance row
  float4 pc = xs[ln];                   // software-pipelined candidate tile
  for (int jb = 0; jb < NPTS; jb += 16) {
    v2f bv; bv.x = hf ? pc.z : pc.x; bv.y = hf ? 0.f : pc.y;
    const float base = pc.w;            // ||x_j||^2 for column ln
    v8f c0 = {}; v8f c1 = {};
    c0 = wmma4(a0, bv, c0);             // dot(query 0..15,  cand 0..15)
    c1 = wmma4(a1, bv, c1);             // dot(query 16..31, cand 0..15)
#pragma unroll
    for (int v = 0; v < 8; ++v) {       // stash (sqJ - 2*dot); owner adds sqI
      dtile[w][v + 8 * hf][ln]      = fmaf(-2.f, c0[v], base);
      dtile[w][16 + v + 8 * hf][ln] = fmaf(-2.f, c1[v], base);
    }
    __builtin_amdgcn_wave_barrier();    // same-wave LDS is in-order (DScnt)

    // batched row fetch: 4x ds_load_b128 instead of 16 scalar round-trips
    float4 r0 = *reinterpret_cast<const float4*>(rowp + 0);
    float4 r1 = *reinterpret_cast<const float4*>(rowp + 4);
    float4 r2 = *reinterpret_cast<const float4*>(rowp + 8);
    float4 r3 = *reinterpret_cast<const float4*>(rowp + 12);
    // prefetch next chunk's candidate float4 (overlaps with selection below)
    const int jn = (jb + 16 < NPTS) ? (jb + 16) : 0;
    pc = xs[jn + ln];

    const float dv[16] = {r0.x, r0.y, r0.z, r0.w, r1.x, r1.y, r1.z, r1.w,
                          r2.x, r2.y, r2.z, r2.w, r3.x, r3.y, r3.z, r3.w};
#pragma unroll
    for (int c = 0; c < 16; ++c) {
      float d2 = sqI + dv[c];
      int   j  = jb + c;
      if (d2 < curMax) {                // rare: ~140 inserts per 8192 candidates
#pragma unroll
        for (int t = 0; t < KNN; ++t) if (t == maxSlot) { kd[t] = d2; ki[t] = j; }
        curMax = kd[0]; maxSlot = 0;
#pragma unroll
        for (int t = 1; t < KNN; ++t) if (kd[t] > curMax) { curMax = kd[t]; maxSlot = t; }
      }
    }
  }

  // emit ascending-by-distance neighbor list (matches top_k ordering)
  const int q = qw + l;
  int* outp = knn_idx + ((size_t)b * NPTS + q) * KNN;
  for (int r = 0; r < KNN; ++r) {
    float best = kd[0]; int bestj = ki[0]; int bs = 0;
#pragma unroll
    for (int t = 1; t < KNN; ++t) if (kd[t] < best) { best = kd[t]; bestj = ki[t]; bs = t; }
    outp[r] = bestj;
#pragma unroll
    for (int t = 0; t < KNN; ++t) if (t == bs) kd[t] = 3.0e38f;
  }
}

// ---------------------------------------------------------------- kernels 3 & 5
// Per wave: 16 edges x 64 channels = 8 WMMAs (4 N-tiles x 2 K-steps, K 6->8).
template <bool STATS>
__global__ __launch_bounds__(256) void edge_kernel(const float* __restrict__ x,
                                                   const float* __restrict__ W,
                                                   const int* __restrict__ knn_idx,
                                                   float* __restrict__ stats,
                                                   float* __restrict__ out) {
  __shared__ float hbuf[8][16][68];     // padded: conflict-free transpose

  const int w  = threadIdx.x >> 5;
  const int l  = threadIdx.x & 31;
  const int hf = l >> 4;
  const int ln = l & 15;

  // loop-invariant B operands from W (O=64 x E=6, row-major), zero-padded K
  v2f bw[4][2];
#pragma unroll
  for (int t = 0; t < 4; ++t)
#pragma unroll
    for (int s = 0; s < 2; ++s) {
      int o  = 16 * t + ln;
      int k0 = 4 * s + 2 * hf;
      bw[t][s].x = (k0     < 6) ? W[o * 6 + k0]     : 0.f;
      bw[t][s].y = (k0 + 1 < 6) ? W[o * 6 + k0 + 1] : 0.f;
    }

  float sc0 = 0.f, sh0 = 0.f, sc1 = 0.f, sh1 = 0.f;
  if (!STATS) {
    sc0 = stats[128 + l];      sh0 = stats[192 + l];
    sc1 = stats[128 + 32 + l]; sh1 = stats[192 + 32 + l];
  }
  float sA0 = 0.f, sA1 = 0.f, sQ0 = 0.f, sQ1 = 0.f;

  const int nGroups = TOTE / 16;                 // 40960 groups of 16 edges
  const int waveId  = blockIdx.x * 8 + w;
  const int nWaves  = gridDim.x * 8;

  for (int g = waveId; g < nGroups; g += nWaves) {
    const int ebase = g * 16;
    const int e = ebase + ln;                    // lane's A-row edge (m = ln)
    const int b = e / EPB;
    const int r = e - b * EPB;
    const int n = r / KNN;
    const int j = knn_idx[e];
    const float* xi = x + ((size_t)b * NPTS + n) * 3;
    const float* xj = x + ((size_t)b * NPTS + j) * 3;
    float e0 = xi[0], e1 = xi[1], e2 = xi[2];
    float e3 = xj[0] - e0, e4 = xj[1] - e1, e5 = xj[2] - e2;
    // A operand per K-step: lane supplies K = 4s + 2*hf + {0,1}
    v2f a0; a0.x = hf ? e2 : e0;  a0.y = hf ? e3  : e1;
    v2f a1; a1.x = hf ? 0.f : e4; a1.y = hf ? 0.f : e5;

    v8f c0 = {}, c1 = {}, c2 = {}, c3 = {};
    c0 = wmma4(a0, bw[0][0], c0); c0 = wmma4(a1, bw[0][1], c0);
    c1 = wmma4(a0, bw[1][0], c1); c1 = wmma4(a1, bw[1][1], c1);
    c2 = wmma4(a0, bw[2][0], c2); c2 = wmma4(a1, bw[2][1], c2);
    c3 = wmma4(a0, bw[3][0], c3); c3 = wmma4(a1, bw[3][1], c3);

#pragma unroll
    for (int v = 0; v < 8; ++v) {
      int m = v + 8 * hf;
      hbuf[w][m][0 * 16 + ln] = c0[v];
      hbuf[w][m][1 * 16 + ln] = c1[v];
      hbuf[w][m][2 * 16 + ln] = c2[v];
      hbuf[w][m][3 * 16 + ln] = c3[v];
    }
    __builtin_amdgcn_wave_barrier();

    if (STATS) {
#pragma unroll
      for (int m = 0; m < 16; ++m) {
        float h0 = hbuf[w][m][l];
        float h1 = hbuf[w][m][32 + l];
        sA0 += h0; sQ0 = fmaf(h0, h0, sQ0);
        sA1 += h1; sQ1 = fmaf(h1, h1, sQ1);
      }
    } else {
      const int bb    = ebase / EPB;
      const int rbase = ebase - bb * EPB;        // EPB % 16 == 0: no straddle
#pragma unroll
      for (int p = 0; p < 2; ++p) {
        int   o  = l + 32 * p;
        float sc = p ? sc1 : sc0;
        float sh = p ? sh1 : sh0;
        float* dst = out + ((size_t)(bb * NOUT + o)) * EPB + rbase;
#pragma unroll
        for (int u = 0; u < 4; ++u) {            // 16 contiguous floats / lane
          float4 v4;
          v4.x = fmaxf(fmaf(hbuf[w][4 * u + 0][o], sc, sh), 0.f);
          v4.y = fmaxf(fmaf(hbuf[w][4 * u + 1][o], sc, sh), 0.f);
          v4.z = fmaxf(fmaf(hbuf[w][4 * u + 2][o], sc, sh), 0.f);
          v4.w = fmaxf(fmaf(hbuf[w][4 * u + 3][o], sc, sh), 0.f);
          *reinterpret_cast<float4*>(dst + 4 * u) = v4;
        }
      }
    }
    __builtin_amdgcn_wave_barrier();
  }

  if (STATS) {
    atomicAdd(&stats[l],           sA0);
    atomicAdd(&stats[l + 32],      sA1);
    atomicAdd(&stats[64 + l],      sQ0);
    atomicAdd(&stats[64 + 32 + l], sQ1);
  }
}

// ---------------------------------------------------------------- kernel 4
__global__ void finalize_kernel(const float* __restrict__ gamma,
                                const float* __restrict__ beta,
                                float* __restrict__ stats) {
  int o = threadIdx.x;
  if (o < NOUT) {
    const float inv  = 1.0f / (float)TOTE;
    float mean = stats[o] * inv;
    float var  = fmaf(-mean, mean, stats[64 + o] * inv);
    float sc   = gamma[o] * rsqrtf(var + 1e-5f);
    stats[128 + o] = sc;
    stats[192 + o] = fmaf(-mean, sc, beta[o]);
  }
}

// ---------------------------------------------------------------- launcher
extern "C" void kernel_launch(void* const* d_in, const int* in_sizes, int n_in,
                              void* d_out, int out_size, void* d_ws, size_t ws_size,
                              hipStream_t stream) {
  (void)in_sizes; (void)n_in; (void)out_size; (void)ws_size;
  const float* x     = (const float*)d_in[0];
  const float* W     = (const float*)d_in[1];
  const float* gamma = (const float*)d_in[2];
  const float* beta  = (const float*)d_in[3];
  // d_in[4] is k (==20), baked in as KNN
  float* out = (float*)d_out;

  float* sq      = (float*)d_ws;                                     // 32768 f
  int*   knn_idx = (int*)((char*)d_ws + 131072);                     // 655360 i
  float* stats   = (float*)((char*)d_ws + 131072 + 2621440);         // 256 f

  sq_kernel<<<128, 256, 0, stream>>>(x, sq, stats);
  knn_kernel<<<NB * (NPTS / 256), 256, 0, stream>>>(x, sq, knn_idx);
  edge_kernel<true><<<320, 256, 0, stream>>>(x, W, knn_idx, stats, out);
  finalize_kernel<<<1, 64, 0, stream>>>(gamma, beta, stats);
  edge_kernel<false><<<320, 256, 0, stream>>>(x, W, knn_idx, stats, out);
}